// HypergraphRayleighQuotientLossDirect_75651553952122
// MI455X (gfx1250) — compile-verified
//
#include <hip/hip_runtime.h>
#include <hip/hip_bf16.h>

typedef __attribute__((ext_vector_type(2))) float v2f;
typedef __attribute__((ext_vector_type(8))) float v8f;

#define KDIM 32
#define EPS 1e-8f

// ---------------------------------------------------------------------------
// Block-wide reduction of a 32-wide float vector (lane = column) across the
// 8 waves of a 256-thread block, using V_WMMA_F32_16X16X4_F32.
//
// Each wave deposits its 32-wide accumulator into LDS (8 rows x 32 cols).
// Wave 0 then builds B operands (4 rows of 16 columns per WMMA, per the ISA
// rule "B rows striped across lanes within one VGPR": VGPR0 holds rows
// {0,2} in lane halves, VGPR1 rows {1,3}) and multiplies by an all-ones A.
// Every element of D is then the column sum of B's 4 rows; chaining C over
// two WMMAs folds all 8 wave-rows. Column sums are invariant to any K-row
// permutation, so only the lane->N mapping matters. Result is atomically
// added to gout[0..31].
// ---------------------------------------------------------------------------
__device__ __forceinline__ void block_reduce32_wmma(float val, float* gout) {
  __shared__ float red[8][KDIM];
  const int lane = threadIdx.x & 31;
  const int wave = threadIdx.x >> 5;
  red[wave][lane] = val;
  __syncthreads();
  if (wave == 0) {
    const int rbase = (lane < 16) ? 0 : 2;   // lane half -> K rows {0,1} / {2,3}
    const int c0 = (lane & 15);              // columns 0..15
    const int c1 = 16 + (lane & 15);         // columns 16..31

    v2f a; a.x = 1.0f; a.y = 1.0f;           // A = ones(16x4)

    // Columns 0..15: rows 0-3 then rows 4-7 (C-chained)
    v2f b0, b1;
    b0.x = red[rbase + 0][c0]; b0.y = red[rbase + 1][c0];
    b1.x = red[rbase + 4][c0]; b1.y = red[rbase + 5][c0];
    v8f acc0 = {};
    acc0 = __builtin_amdgcn_wmma_f32_16x16x4_f32(false, a, false, b0, (short)0,
                                                 acc0, false, false);
    acc0 = __builtin_amdgcn_wmma_f32_16x16x4_f32(false, a, false, b1, (short)0,
                                                 acc0, false, false);

    // Columns 16..31
    v2f b2, b3;
    b2.x = red[rbase + 0][c1]; b2.y = red[rbase + 1][c1];
    b3.x = red[rbase + 4][c1]; b3.y = red[rbase + 5][c1];
    v8f acc1 = {};
    acc1 = __builtin_amdgcn_wmma_f32_16x16x4_f32(false, a, false, b2, (short)0,
                                                 acc1, false, false);
    acc1 = __builtin_amdgcn_wmma_f32_16x16x4_f32(false, a, false, b3, (short)0,
                                                 acc1, false, false);

    // D row M=0 lives in VGPR0 (lanes 0-15, N = lane). Every row equals the
    // column sum, so lanes 0..15 carry the 16 sums for each half.
    if (lane < 16) {
      atomicAdd(&gout[c0], acc0[0]);
      atomicAdd(&gout[c1], acc1[0]);
    }
  }
  __syncthreads();
}

// ---------------------------------------------------------------------------
// K1: Dv[v] += w[e];  De[e] += 1   (one thread per nnz entry, grid-stride)
// ---------------------------------------------------------------------------
__global__ void degrees_kernel(const int* __restrict__ nidx,
                               const int* __restrict__ eidx,
                               const float* __restrict__ hw,
                               float* __restrict__ Dv,
                               float* __restrict__ De, int nnz) {
  int i = blockIdx.x * blockDim.x + threadIdx.x;
  const int stride = gridDim.x * blockDim.x;
  for (; i < nnz; i += stride) {
    const int e = eidx[i];
    const int v = nidx[i];
    atomicAdd(&Dv[v], hw[e]);
    atomicAdd(&De[e], 1.0f);
  }
}

// ---------------------------------------------------------------------------
// K2: normZ[v,k] = Dv^-1/2 * Z[v,k];  f_Dv_f[k] += Z[v,k]^2 * Dv[v]
//     (one wave per node, lane = column; WMMA block reduction)
// ---------------------------------------------------------------------------
__global__ void normz_fdvf_kernel(const float* __restrict__ Z,
                                  const float* __restrict__ Dv,
                                  float* __restrict__ normZ,
                                  float* __restrict__ fdvf, int N) {
  const int lane = threadIdx.x & 31;
  const int wib = threadIdx.x >> 5;
  const int wpb = blockDim.x >> 5;
  const int gwave = blockIdx.x * wpb + wib;
  const int nwaves = gridDim.x * wpb;
  float acc = 0.0f;
  for (int v = gwave; v < N; v += nwaves) {
    float dv = Dv[v];
    dv = (dv == 0.0f) ? 1.0f : dv;
    const float s = 1.0f / sqrtf(dv);
    const float z = Z[v * KDIM + lane];
    normZ[v * KDIM + lane] = z * s;
    acc = fmaf(z * z, dv, acc);
  }
  block_reduce32_wmma(acc, fdvf);
}

// ---------------------------------------------------------------------------
// K3: per_edge[e,:] += normZ[node,:]   (one wave per nnz entry, lane = column;
//     128B coalesced row read + 128B L2 atomic row; prefetch next row)
// ---------------------------------------------------------------------------
__global__ void scatter_per_edge_kernel(const float* __restrict__ normZ,
                                        const int* __restrict__ nidx,
                                        const int* __restrict__ eidx,
                                        float* __restrict__ per_edge, int nnz) {
  const int lane = threadIdx.x & 31;
  const int wib = threadIdx.x >> 5;
  const int wpb = blockDim.x >> 5;
  const int gwave = blockIdx.x * wpb + wib;
  const int nwaves = gridDim.x * wpb;
  for (int i = gwave; i < nnz; i += nwaves) {
    const int v = nidx[i];
    const int e = eidx[i];
    const int j = i + nwaves;
    if (j < nnz) {
      const int vn = nidx[j];
      __builtin_prefetch(&normZ[vn * KDIM + lane], 0, 0);  // global_prefetch_b8
    }
    const float val = normZ[v * KDIM + lane];
    atomicAdd(&per_edge[e * KDIM + lane], val);
  }
}

// ---------------------------------------------------------------------------
// K4: theta[k] += (w[e]/De[e]) * per_edge[e,k]^2   (wave per edge, lane = k)
// ---------------------------------------------------------------------------
__global__ void theta_kernel(const float* __restrict__ per_edge,
                             const float* __restrict__ hw,
                             const float* __restrict__ De,
                             float* __restrict__ theta, int E) {
  const int lane = threadIdx.x & 31;
  const int wib = threadIdx.x >> 5;
  const int wpb = blockDim.x >> 5;
  const int gwave = blockIdx.x * wpb + wib;
  const int nwaves = gridDim.x * wpb;
  float acc = 0.0f;
  for (int e = gwave; e < E; e += nwaves) {
    float de = De[e];
    de = (de == 0.0f) ? 1.0f : de;
    const float coeff = hw[e] / de;
    const float s = per_edge[e * KDIM + lane];
    acc = fmaf(coeff * s, s, acc);
  }
  block_reduce32_wmma(acc, theta);
}

// ---------------------------------------------------------------------------
// K5: out = mean_k (1 - theta[k] / (f_Dv_f[k] + EPS))   (single wave32)
// ---------------------------------------------------------------------------
__global__ void finalize_kernel(const float* __restrict__ theta,
                                const float* __restrict__ fdvf,
                                float* __restrict__ out) {
  const int lane = threadIdx.x;
  float r = 1.0f - theta[lane] / (fdvf[lane] + EPS);
  for (int off = 16; off > 0; off >>= 1) r += __shfl_xor(r, off, 32);
  if (lane == 0) out[0] = r * (1.0f / 32.0f);
}

extern "C" void kernel_launch(void* const* d_in, const int* in_sizes, int n_in,
                              void* d_out, int out_size, void* d_ws,
                              size_t ws_size, hipStream_t stream) {
  const float* Z = (const float*)d_in[0];
  const int* nidx = (const int*)d_in[1];
  const int* eidx = (const int*)d_in[2];
  const float* hw = (const float*)d_in[3];

  const int N = in_sizes[0] / KDIM;   // num_nodes      (100000)
  const int nnz = in_sizes[1];        // nnz            (3200000)
  const int E = in_sizes[3];          // num_hyperedges (200000)

  // Workspace layout (floats): Dv | De | theta(32) | fdvf(32) | per_edge | normZ
  float* ws = (float*)d_ws;
  float* Dv = ws;
  float* De = Dv + N;
  float* theta = De + E;
  float* fdvf = theta + KDIM;
  float* per_edge = fdvf + KDIM;
  float* normZ = per_edge + (size_t)E * KDIM;

  // Zero everything that is accumulated into (normZ is fully overwritten).
  const size_t zero_bytes =
      ((size_t)N + (size_t)E + 2 * KDIM + (size_t)E * KDIM) * sizeof(float);
  hipMemsetAsync(d_ws, 0, zero_bytes, stream);

  const int threads = 256;

  degrees_kernel<<<4096, threads, 0, stream>>>(nidx, eidx, hw, Dv, De, nnz);
  normz_fdvf_kernel<<<2048, threads, 0, stream>>>(Z, Dv, normZ, fdvf, N);
  scatter_per_edge_kernel<<<8192, threads, 0, stream>>>(normZ, nidx, eidx,
                                                        per_edge, nnz);
  theta_kernel<<<2048, threads, 0, stream>>>(per_edge, hw, De, theta, E);
  finalize_kernel<<<1, 32, 0, stream>>>(theta, fdvf, (float*)d_out);
}